// EdgesFeatures_77833397338254
// MI455X (gfx1250) — compile-verified
//
#include <hip/hip_runtime.h>
#include <hip/hip_bf16.h>

// ---------------------------------------------------------------------------
// Problem constants (match reference)
// ---------------------------------------------------------------------------
#define BB      32      // batch
#define SS      64      // S
#define DD      8       // D
#define NEE     512     // NE = S*D
#define IN0     16
#define IN1     32
#define IN2     32
#define OUT0    32
#define OUT1    32
#define OUT2    16

typedef __attribute__((ext_vector_type(2))) float v2f;
typedef __attribute__((ext_vector_type(8))) float v8f;

static __device__ __forceinline__ v8f wmma_f32_4(v2f a, v2f b, v8f c) {
    // D = A(16x4 f32) * B(4x16 f32) + C(16x16 f32)
    return __builtin_amdgcn_wmma_f32_16x16x4_f32(
        /*neg_a=*/false, a, /*neg_b=*/false, b,
        /*c_mod=*/(short)0, c, /*reuse_a=*/false, /*reuse_b=*/false);
}

// ---------------------------------------------------------------------------
// Binning: group edges by eidx = c2*D + c3  (512 bins)
// ---------------------------------------------------------------------------
__global__ void zero_bins_kernel(int* __restrict__ bins) {
    int i = blockIdx.x * blockDim.x + threadIdx.x;
    if (i < NEE) bins[i] = 0;
}

__global__ void hist_kernel(const int* __restrict__ inc, int* __restrict__ bins, int E) {
    int i = blockIdx.x * blockDim.x + threadIdx.x;
    if (i < E) {
        int eidx = inc[i * 4 + 2] * DD + inc[i * 4 + 3];
        atomicAdd(&bins[eidx], 1);
    }
}

__global__ __launch_bounds__(NEE) void scan_kernel(const int* __restrict__ bins,
                                                   int* __restrict__ offsets,
                                                   int* __restrict__ cursor) {
    __shared__ int s[NEE];
    int t = threadIdx.x;
    int mine = bins[t];
    s[t] = mine;
    __syncthreads();
    // Hillis-Steele inclusive scan over 512 entries
    for (int d = 1; d < NEE; d <<= 1) {
        int v = (t >= d) ? s[t - d] : 0;
        __syncthreads();
        s[t] += v;
        __syncthreads();
    }
    int excl = s[t] - mine;
    offsets[t] = excl;
    cursor[t]  = excl;
}

__global__ void scatter_kernel(const int* __restrict__ inc, int* __restrict__ cursor,
                               int* __restrict__ order, int E) {
    int i = blockIdx.x * blockDim.x + threadIdx.x;
    if (i < E) {
        int eidx = inc[i * 4 + 2] * DD + inc[i * 4 + 3];
        int p = atomicAdd(&cursor[eidx], 1);
        order[p] = i;
    }
}

// ---------------------------------------------------------------------------
// Stage A: one block (1 wave) per eidx. GEMM over tiles of 16 edges:
//   H(32x16) = relu(W0(32x16) * X(16x16e) + b0), Y(32x16) = relu(W1(32x32)*H + b1)
// then mean-accumulate per c0 into lsum (LDS), flush to sums[e][c0][o].
// ---------------------------------------------------------------------------
__global__ __launch_bounds__(32) void stageA_kernel(
    const float* __restrict__ nodes,   // (NNODES*S, 16)
    const int*   __restrict__ inc,     // (E,4)
    const float* __restrict__ W0,      // (NE,32,16)
    const float* __restrict__ b0,      // (NE,32)
    const float* __restrict__ W1,      // (NE,32,32)
    const float* __restrict__ b1,      // (NE,32)
    const int*   __restrict__ order,
    const int*   __restrict__ bins,
    const int*   __restrict__ offsets,
    float* __restrict__ sums,          // (NE, 32(c0), 32(o))
    float* __restrict__ cnt)           // (NE, 32(c0))
{
    const int e    = blockIdx.x;       // eidx
    const int lane = threadIdx.x;      // 0..31 (wave32)
    const int row   = lane & 15;
    const int hi    = lane >> 4;       // 0/1: which half of lanes
    const int khalf = hi * 2;
    const int n     = lane & 15;       // column (edge within tile)

    __shared__ float xs[16][17];       // X staged:   xs[edge][feature]
    __shared__ float hs[32][17];       // H staged:   hs[o][edge]
    __shared__ float lsum[BB][OUT0];   // per-c0 accumulators
    __shared__ float lcnt[BB];
    __shared__ float bs0[32], bs1[32];
    __shared__ int   c0s[16];

    for (int i = lane; i < BB * OUT0; i += 32) ((float*)lsum)[i] = 0.0f;
    lcnt[lane] = 0.0f;
    bs0[lane] = b0[e * 32 + lane];
    bs1[lane] = b1[e * 32 + lane];

    const int start = offsets[e];
    const int count = bins[e];

    // Preload A-matrix fragments (weights) for both layers.
    const float* W0e = W0 + e * (32 * 16);
    const float* W1e = W1 + e * (32 * 32);
    v2f a0[2][4];                      // W0: M tiles {0,1}, K steps 4 (K=16)
    for (int m = 0; m < 2; ++m)
        for (int kk = 0; kk < 4; ++kk) {
            int o = m * 16 + row, k = kk * 4 + khalf;
            v2f a; a.x = W0e[o * 16 + k]; a.y = W0e[o * 16 + k + 1];
            a0[m][kk] = a;
        }
    v2f a1[2][8];                      // W1: M tiles {0,1}, K steps 8 (K=32)
    for (int m = 0; m < 2; ++m)
        for (int kk = 0; kk < 8; ++kk) {
            int o = m * 16 + row, k = kk * 4 + khalf;
            v2f a; a.x = W1e[o * 32 + k]; a.y = W1e[o * 32 + k + 1];
            a1[m][kk] = a;
        }
    __syncthreads();

    for (int t = 0; t < count; t += 16) {
        const int nvalid = count - t < 16 ? count - t : 16;
        // ---- gather 16 edges' features into LDS (each lane loads 8 floats)
        const int fbase = hi * 8;
        if (n < nvalid) {
            int eg  = order[start + t + n];
            int ec0 = inc[eg * 4 + 0];
            int ec1 = inc[eg * 4 + 1];
            int ec2 = inc[eg * 4 + 2];
            if (lane < 16) { c0s[n] = ec0; atomicAdd(&lcnt[ec0], 1.0f); }
            const float* xp = nodes + (ec1 * SS + ec2) * IN0;
            #pragma unroll
            for (int f = 0; f < 8; ++f) xs[n][fbase + f] = xp[fbase + f];
        } else {
            if (lane < 16) c0s[n] = -1;
            #pragma unroll
            for (int f = 0; f < 8; ++f) xs[n][fbase + f] = 0.0f;
        }
        __syncthreads();

        // ---- layer 0: H = W0 * X   (M=32, N=16, K=16)
        v8f h0 = {}; v8f h1 = {};
        #pragma unroll
        for (int kk = 0; kk < 4; ++kk) {
            int k = kk * 4 + khalf;
            v2f bf; bf.x = xs[n][k]; bf.y = xs[n][k + 1];
            h0 = wmma_f32_4(a0[0][kk], bf, h0);
            h1 = wmma_f32_4(a0[1][kk], bf, h1);
        }
        // bias + relu, stage H into LDS as hs[o][edge]
        #pragma unroll
        for (int v = 0; v < 8; ++v) {
            int o = v + 8 * hi;
            float z0 = h0[v] + bs0[o];      hs[o][n]      = z0 > 0.0f ? z0 : 0.0f;
            float z1 = h1[v] + bs0[16 + o]; hs[16 + o][n] = z1 > 0.0f ? z1 : 0.0f;
        }
        __syncthreads();

        // ---- layer 1: Y = W1 * H   (M=32, N=16, K=32)
        v8f y0 = {}; v8f y1 = {};
        #pragma unroll
        for (int kk = 0; kk < 8; ++kk) {
            int k = kk * 4 + khalf;
            v2f bf; bf.x = hs[k][n]; bf.y = hs[k + 1][n];
            y0 = wmma_f32_4(a1[0][kk], bf, y0);
            y1 = wmma_f32_4(a1[1][kk], bf, y1);
        }
        // bias + relu + segment-accumulate by c0 (LDS float atomics)
        const int ec0 = c0s[n];
        if (ec0 >= 0) {
            #pragma unroll
            for (int v = 0; v < 8; ++v) {
                int o = v + 8 * hi;
                float z0 = y0[v] + bs1[o];      z0 = z0 > 0.0f ? z0 : 0.0f;
                float z1 = y1[v] + bs1[16 + o]; z1 = z1 > 0.0f ? z1 : 0.0f;
                atomicAdd(&lsum[ec0][o], z0);
                atomicAdd(&lsum[ec0][16 + o], z1);
            }
        }
        __syncthreads();
    }

    // Flush: segment c0*NE+e is owned exclusively by this block -> plain stores
    for (int i = lane; i < BB * OUT0; i += 32) {
        int c0i = i >> 5, o = i & 31;
        sums[(e * BB + c0i) * OUT0 + o] = lsum[c0i][o];
    }
    cnt[e * BB + lane] = lcnt[lane];
}

// ---------------------------------------------------------------------------
// Stage B: one block (1 wave) per eidx.
//   EF(32b x 32i) = sums/cnt ; Z = relu(Wo0(32x32)*EF^T + bo0)
//   Y = relu(Wo1(16x32)*Z + bo1) ; out[(b*NE+e)*16 + o] = Y[o][b]
// ---------------------------------------------------------------------------
__global__ __launch_bounds__(32) void stageB_kernel(
    const float* __restrict__ sums,    // (NE,32(c0),32(o))
    const float* __restrict__ cnt,     // (NE,32)
    const float* __restrict__ Wo0,     // (NE,32,32)
    const float* __restrict__ bo0,     // (NE,32)
    const float* __restrict__ Wo1,     // (NE,16,32)
    const float* __restrict__ bo1,     // (NE,16)
    float* __restrict__ out)           // (B*NE,16)
{
    const int e    = blockIdx.x;
    const int lane = threadIdx.x;
    const int row   = lane & 15;
    const int hi    = lane >> 4;
    const int khalf = hi * 2;
    const int n     = lane & 15;

    __shared__ float efs[32][33];      // efs[b][i]
    __shared__ float zs[32][33];       // zs[o][b]
    __shared__ float bs0[32], bs1[16];

    bs0[lane] = bo0[e * 32 + lane];
    if (lane < 16) bs1[lane] = bo1[e * 16 + lane];

    // segment mean (each lane owns batch b = lane)
    {
        const int b = lane;
        float c = cnt[e * BB + b];
        float invc = c > 0.0f ? 1.0f / c : 0.0f;
        const float* sp = sums + (e * BB + b) * OUT0;
        #pragma unroll
        for (int i = 0; i < 32; ++i) efs[b][i] = sp[i] * invc;
    }
    __syncthreads();

    const float* W0e = Wo0 + e * 32 * 32;
    const float* W1e = Wo1 + e * 16 * 32;

    // ---- layer 0: M=32 (2 tiles), N=32 (2 tiles), K=32 (8 steps)
    v8f acc[2][2] = {};
    #pragma unroll
    for (int kk = 0; kk < 8; ++kk) {
        int k = kk * 4 + khalf;
        v2f bf0; bf0.x = efs[n][k];      bf0.y = efs[n][k + 1];       // cols b=0..15
        v2f bf1; bf1.x = efs[16 + n][k]; bf1.y = efs[16 + n][k + 1];  // cols b=16..31
        #pragma unroll
        for (int m = 0; m < 2; ++m) {
            int o = m * 16 + row;
            v2f af; af.x = W0e[o * 32 + k]; af.y = W0e[o * 32 + k + 1];
            acc[m][0] = wmma_f32_4(af, bf0, acc[m][0]);
            acc[m][1] = wmma_f32_4(af, bf1, acc[m][1]);
        }
    }
    #pragma unroll
    for (int m = 0; m < 2; ++m)
        #pragma unroll
        for (int nt = 0; nt < 2; ++nt)
            #pragma unroll
            for (int v = 0; v < 8; ++v) {
                int o = m * 16 + v + 8 * hi;
                int b = nt * 16 + n;
                float z = acc[m][nt][v] + bs0[o];
                zs[o][b] = z > 0.0f ? z : 0.0f;
            }
    __syncthreads();

    // ---- layer 1: M=16, N=32 (2 tiles), K=32 (8 steps)
    v8f y[2] = {};
    #pragma unroll
    for (int kk = 0; kk < 8; ++kk) {
        int k = kk * 4 + khalf;
        v2f af; af.x = W1e[row * 32 + k]; af.y = W1e[row * 32 + k + 1];
        v2f bf0; bf0.x = zs[k][n];      bf0.y = zs[k + 1][n];
        v2f bf1; bf1.x = zs[k][16 + n]; bf1.y = zs[k + 1][16 + n];
        y[0] = wmma_f32_4(af, bf0, y[0]);
        y[1] = wmma_f32_4(af, bf1, y[1]);
    }
    #pragma unroll
    for (int nt = 0; nt < 2; ++nt) {
        int b = nt * 16 + n;
        float* op = out + (b * NEE + e) * OUT2;
        #pragma unroll
        for (int v = 0; v < 8; ++v) {
            int o = v + 8 * hi;
            float z = y[nt][v] + bs1[o];
            op[o] = z > 0.0f ? z : 0.0f;
        }
    }
}

// ---------------------------------------------------------------------------
// Host launcher
// ---------------------------------------------------------------------------
extern "C" void kernel_launch(void* const* d_in, const int* in_sizes, int n_in,
                              void* d_out, int out_size, void* d_ws, size_t ws_size,
                              hipStream_t stream) {
    const float* nodes = (const float*)d_in[0];
    const float* W0    = (const float*)d_in[1];
    const float* b0    = (const float*)d_in[2];
    const float* W1    = (const float*)d_in[3];
    const float* b1    = (const float*)d_in[4];
    const float* Wo0   = (const float*)d_in[5];
    const float* bo0   = (const float*)d_in[6];
    const float* Wo1   = (const float*)d_in[7];
    const float* bo1   = (const float*)d_in[8];
    const int*   inc   = (const int*)d_in[9];
    const int E = in_sizes[9] / 4;

    // Workspace layout
    int*   bins    = (int*)d_ws;              // 512
    int*   offsets = bins + NEE;              // 512
    int*   cursor  = offsets + NEE;           // 512
    int*   order   = cursor + NEE;            // E
    size_t fof     = ((size_t)(NEE * 3 + E) * sizeof(int) + 255) & ~(size_t)255;
    float* sums    = (float*)((char*)d_ws + fof);   // NE*32*32
    float* cnt     = sums + NEE * BB * OUT0;        // NE*32

    float* out = (float*)d_out;

    zero_bins_kernel<<<(NEE + 255) / 256, 256, 0, stream>>>(bins);
    hist_kernel<<<(E + 255) / 256, 256, 0, stream>>>(inc, bins, E);
    scan_kernel<<<1, NEE, 0, stream>>>(bins, offsets, cursor);
    scatter_kernel<<<(E + 255) / 256, 256, 0, stream>>>(inc, cursor, order, E);
    stageA_kernel<<<NEE, 32, 0, stream>>>(nodes, inc, W0, b0, W1, b1,
                                          order, bins, offsets, sums, cnt);
    stageB_kernel<<<NEE, 32, 0, stream>>>(sums, cnt, Wo0, bo0, Wo1, bo1, out);
}